// DNN_Beamformer_21620865368255
// MI455X (gfx1250) — compile-verified
//
#include <hip/hip_runtime.h>
#include <math.h>

// MVDR DNN beamformer for MI455X (gfx1250, wave32).
// Heavy op: masked PSD = 16x16xT realified SGEMM per (b,f) via v_wmma_f32_16x16x4_f32,
// mask weight folded into the A operand only (D = (w.*R) x R^T).
// Staging uses GLOBAL_LOAD_ASYNC_TO_LDS (ASYNCcnt) with LDS double buffering.

#define B_  8
#define T_  1024
#define C_  8
#define F_  513
#define A_  320
#define FT  16      // f-tile per workgroup (16 waves, one f per wave)
#define TC  16      // t-chunk staged in LDS (4 wmma K-steps)
#define XS  34      // padded LDS row stride (dwords): conflict-free fragment reads
#define EPS_MASK 1e-6f
#define EPS_PSD  1e-15f
#define EPS_MVDR 1e-15f

typedef float v2f __attribute__((ext_vector_type(2)));
typedef float v8f __attribute__((ext_vector_type(8)));
typedef int   ai2 __attribute__((vector_size(8)));   // matches builtin param type

#if defined(__has_builtin)
#  if __has_builtin(__builtin_amdgcn_global_load_async_to_lds_b64)
#    define HAVE_ASYNC_LDS 1
#  endif
#  if __has_builtin(__builtin_amdgcn_s_wait_asynccnt)
#    define HAVE_WAIT_ASYNC 1
#  endif
#endif

__device__ __forceinline__ void async_b64(const float* g, float* l) {
#ifdef HAVE_ASYNC_LDS
  __builtin_amdgcn_global_load_async_to_lds_b64(
      (__attribute__((address_space(1))) ai2*)(g),
      (__attribute__((address_space(3))) ai2*)(l), 0, 0);
#else
  *(float2*)l = *(const float2*)g;   // sync fallback
#endif
}

template <int N>
__device__ __forceinline__ void wait_async_le() {
#ifdef HAVE_WAIT_ASYNC
  __builtin_amdgcn_s_wait_asynccnt(N);
#else
#ifdef HAVE_ASYNC_LDS
  asm volatile("s_wait_asynccnt %0" ::"i"(N) : "memory");
#endif
#endif
}

// ---------------- Pass 1: msum[b,f,mask] = sum_t mean_c clip(mask) ----------------
__global__ __launch_bounds__(256) void k_mask_sums(const float* __restrict__ ms,
                                                   const float* __restrict__ mn,
                                                   float* __restrict__ msum) {
  int bf = blockIdx.x;                       // b*F + f
  __shared__ float red[2][256];
  const float* bs = ms + (size_t)bf * (C_ * T_);
  const float* bn = mn + (size_t)bf * (C_ * T_);
  float ss = 0.f, sn = 0.f;
  for (int e = threadIdx.x; e < C_ * T_; e += 256) {
    ss += fmaxf(bs[e], EPS_MASK);
    sn += fmaxf(bn[e], EPS_MASK);
  }
  red[0][threadIdx.x] = ss; red[1][threadIdx.x] = sn;
  __syncthreads();
  for (int s = 128; s > 0; s >>= 1) {
    if (threadIdx.x < (unsigned)s) {
      red[0][threadIdx.x] += red[0][threadIdx.x + s];
      red[1][threadIdx.x] += red[1][threadIdx.x + s];
    }
    __syncthreads();
  }
  if (threadIdx.x == 0) {
    msum[bf * 2 + 0] = red[0][0] * 0.125f;   // mean over C=8
    msum[bf * 2 + 1] = red[1][0] * 0.125f;
  }
}

// ---------------- Pass 2: PSD via WMMA, async double-buffered staging ----------------
// R rows 0..7 = Re(x_c), rows 8..15 = Im(x_c).  D = sum_t w_t R[:,t] R[:,t]^T.
// Re(psd[c,e]) = D[c,e] + D[c+8,e+8] ; Im(psd[c,e]) = D[c+8,e] - D[c,e+8].
__global__ __launch_bounds__(512) void k_psd_wmma(const float* __restrict__ x,
    const float* __restrict__ ms, const float* __restrict__ mn,
    const float* __restrict__ msum,
    float* __restrict__ psd_s, float* __restrict__ psd_n) {
  __shared__ float lx[2][TC * C_ * XS];      // 2 x 4352 dwords
  __shared__ float lw[2][2][FT][TC];         // [buf][mask][f][t]
  const int b   = blockIdx.y;
  const int f0  = blockIdx.x * FT;
  const int tid = threadIdx.x;
  const int wv  = tid >> 5;                  // wave -> f within tile
  const int lane = tid & 31;
  const int m  = lane & 15;                  // R row
  const int h  = lane >> 4;                  // half (K group)
  const int cc = m & 7;                      // channel
  const int ri = m >> 3;                     // 0=real row, 1=imag row
  const int f  = f0 + wv;
  const bool fullTile = (f0 + FT) <= F_;
  const int NCH = T_ / TC;                   // 64 chunks

  v8f accS = {0,0,0,0,0,0,0,0};
  v8f accN = {0,0,0,0,0,0,0,0};

  // weight-staging role of this thread (one lw value per thread per chunk)
  const int wmsk = tid >> 8;                 // 0..1 (mask select)
  const int wfl  = (tid & 255) >> 4;         // f within tile
  const int wtl  = tid & 15;                 // t within chunk
  const float* wmp = wmsk ? mn : ms;

  if (fullTile) {
    const int ff = f0 + wfl;
    const size_t wbase = ((size_t)(b * F_ + ff) * C_) * T_;
    const float winv = 0.125f / (msum[(b * F_ + ff) * 2 + wmsk] + EPS_PSD);

    // x staging role: 4 b64 segments per thread per chunk
    // prologue: stage chunk 0
    #pragma unroll
    for (int it = 0; it < 4; ++it) {
      int sgi = tid + it * 512;              // 0..2047
      int row = sgi >> 4;                    // tl*8 + c
      int sg  = sgi & 15;
      int tl = row >> 3, c2 = row & 7;
      const float* gp = x + ((size_t)(b * T_ + tl) * C_ + c2) * (F_ * 2) + f0 * 2 + sg * 2;
      async_b64(gp, &lx[0][row * XS + sg * 2]);
    }
    {
      float sum = 0.f;
      #pragma unroll
      for (int c2 = 0; c2 < C_; ++c2) sum += fmaxf(wmp[wbase + (size_t)c2 * T_ + wtl], EPS_MASK);
      lw[0][wmsk][wfl][wtl] = sum * winv;
    }

    for (int i = 0; i < NCH; ++i) {
      const int cur = i & 1;
      const bool pf = (i + 1) < NCH;
      float mv[8];
      if (pf) {
        const int tcg = (i + 1) * TC;
        #pragma unroll
        for (int it = 0; it < 4; ++it) {
          int sgi = tid + it * 512;
          int row = sgi >> 4;
          int sg  = sgi & 15;
          int tl = row >> 3, c2 = row & 7;
          const float* gp = x + ((size_t)(b * T_ + tcg + tl) * C_ + c2) * (F_ * 2) + f0 * 2 + sg * 2;
          async_b64(gp, &lx[cur ^ 1][row * XS + sg * 2]);
        }
        #pragma unroll
        for (int c2 = 0; c2 < C_; ++c2) mv[c2] = wmp[wbase + (size_t)c2 * T_ + tcg + wtl];
      }
      if (pf) wait_async_le<4>(); else wait_async_le<0>();
      __syncthreads();
      // ---- 4 K-steps of 2 WMMAs (speech + noise) on buffer `cur` ----
      #pragma unroll
      for (int s = 0; s < 4; ++s) {
        int ta = s * 4 + 2 * h;              // per-lane K indices (A 16x4 layout)
        int tb = ta + 1;
        float r0 = lx[cur][(ta * C_ + cc) * XS + wv * 2 + ri];
        float r1 = lx[cur][(tb * C_ + cc) * XS + wv * 2 + ri];
        float ws0 = lw[cur][0][wv][ta], ws1 = lw[cur][0][wv][tb];
        float wn0 = lw[cur][1][wv][ta], wn1 = lw[cur][1][wv][tb];
        v2f rf;  rf.x  = r0;        rf.y  = r1;        // B = R
        v2f as_; as_.x = ws0 * r0;  as_.y = ws1 * r1;  // A = w_s .* R
        v2f an_; an_.x = wn0 * r0;  an_.y = wn1 * r1;  // A = w_n .* R
        accS = __builtin_amdgcn_wmma_f32_16x16x4_f32(false, as_, false, rf, (short)0, accS, false, false);
        accN = __builtin_amdgcn_wmma_f32_16x16x4_f32(false, an_, false, rf, (short)0, accN, false, false);
      }
      if (pf) {
        float sum = 0.f;
        #pragma unroll
        for (int c2 = 0; c2 < C_; ++c2) sum += fmaxf(mv[c2], EPS_MASK);
        lw[cur ^ 1][wmsk][wfl][wtl] = sum * winv;
      }
      __syncthreads();
    }
  } else {
    // ---- tail tile (single valid f): simple synchronous path ----
    const int ffw = (f0 + wfl > F_ - 1) ? (F_ - 1) : (f0 + wfl);
    const size_t wbase = ((size_t)(b * F_ + ffw) * C_) * T_;
    const float winv = 0.125f / (msum[(b * F_ + ffw) * 2 + wmsk] + EPS_PSD);
    for (int i = 0; i < NCH; ++i) {
      const int tcg = i * TC;
      #pragma unroll
      for (int it = 0; it < 8; ++it) {
        int e = tid + it * 512;              // 4096 floats
        int row = e >> 5;
        int col = e & 31;
        int fl = col >> 1, rr = col & 1;
        int tl = row >> 3, c2 = row & 7;
        int ff = f0 + fl; if (ff > F_ - 1) ff = F_ - 1;
        lx[0][row * XS + col] =
            x[((size_t)(b * T_ + tcg + tl) * C_ + c2) * (F_ * 2) + ff * 2 + rr];
      }
      {
        float sum = 0.f;
        #pragma unroll
        for (int c2 = 0; c2 < C_; ++c2) sum += fmaxf(wmp[wbase + (size_t)c2 * T_ + tcg + wtl], EPS_MASK);
        lw[0][wmsk][wfl][wtl] = sum * winv;
      }
      __syncthreads();
      #pragma unroll
      for (int s = 0; s < 4; ++s) {
        int ta = s * 4 + 2 * h;
        int tb = ta + 1;
        float r0 = lx[0][(ta * C_ + cc) * XS + wv * 2 + ri];
        float r1 = lx[0][(tb * C_ + cc) * XS + wv * 2 + ri];
        float ws0 = lw[0][0][wv][ta], ws1 = lw[0][0][wv][tb];
        float wn0 = lw[0][1][wv][ta], wn1 = lw[0][1][wv][tb];
        v2f rf;  rf.x  = r0;        rf.y  = r1;
        v2f as_; as_.x = ws0 * r0;  as_.y = ws1 * r1;
        v2f an_; an_.x = wn0 * r0;  an_.y = wn1 * r1;
        accS = __builtin_amdgcn_wmma_f32_16x16x4_f32(false, as_, false, rf, (short)0, accS, false, false);
        accN = __builtin_amdgcn_wmma_f32_16x16x4_f32(false, an_, false, rf, (short)0, accN, false, false);
      }
      __syncthreads();
    }
  }

  // ---- combine realified D blocks into complex psd, write out ----
  float* scr = &lx[0][0];                    // flat scratch (8704 dwords >= 16*512)
  const size_t obase = ((size_t)(b * F_ + f)) * (C_ * C_ * 2);
  // speech
  #pragma unroll
  for (int v = 0; v < 8; ++v) scr[wv * 512 + v * 32 + lane] = accS[v];
  __syncthreads();
  if (f < F_) {
    for (int p = lane; p < 64; p += 32) {
      int oc = p >> 3, oe = p & 7;
      float re = scr[wv * 512 + oc * 32 + oe]      + scr[wv * 512 + oc * 32 + 24 + oe];
      float im = scr[wv * 512 + oc * 32 + 16 + oe] - scr[wv * 512 + oc * 32 + 8 + oe];
      *(float2*)(psd_s + obase + oc * 16 + oe * 2) = make_float2(re, im);
    }
  }
  __syncthreads();
  // noise
  #pragma unroll
  for (int v = 0; v < 8; ++v) scr[wv * 512 + v * 32 + lane] = accN[v];
  __syncthreads();
  if (f < F_) {
    for (int p = lane; p < 64; p += 32) {
      int oc = p >> 3, oe = p & 7;
      float re = scr[wv * 512 + oc * 32 + oe]      + scr[wv * 512 + oc * 32 + 24 + oe];
      float im = scr[wv * 512 + oc * 32 + 16 + oe] - scr[wv * 512 + oc * 32 + 8 + oe];
      *(float2*)(psd_n + obase + oc * 16 + oe * 2) = make_float2(re, im);
    }
  }
}

// ---------------- Pass 3: attention logits e[b,c] ----------------
__global__ __launch_bounds__(320) void k_attn(const float* __restrict__ psd_s,
    const float* __restrict__ W, const float* __restrict__ bpsd,
    const float* __restrict__ wg, const float* __restrict__ bg,
    float* __restrict__ elog) {
  __shared__ float feat[F_];
  __shared__ float red[320];
  int b = blockIdx.x >> 3, c = blockIdx.x & 7;
  for (int f = threadIdx.x; f < F_; f += 320) {
    const float* p = psd_s + ((size_t)(b * F_ + f) * C_ + c) * (C_ * 2);
    float sr = 0.f, si = 0.f;
    #pragma unroll
    for (int e = 0; e < C_; ++e) {
      if (e == c) continue;                  // zero diagonal
      sr += p[e * 2]; si += p[e * 2 + 1];
    }
    feat[f] = sqrtf(sr * sr + si * si) * (1.0f / 7.0f);
  }
  __syncthreads();
  int a = threadIdx.x;                       // 0..319
  float acc = 0.f;
  for (int f = 0; f < F_; ++f) acc += feat[f] * W[f * A_ + a];
  red[a] = tanhf(acc + bpsd[a]) * wg[a];
  __syncthreads();
  if (a < 64) red[a] += red[a + 256];
  __syncthreads();
  for (int s = 128; s > 0; s >>= 1) {
    if (a < s) red[a] += red[a + s];
    __syncthreads();
  }
  if (a == 0) elog[b * 8 + c] = red[0] + bg[0];
}

// ---------------- Pass 4: u = softmax(2*e) over channels ----------------
__global__ __launch_bounds__(64) void k_softmax(const float* __restrict__ elog,
                                                float* __restrict__ u) {
  int t = threadIdx.x;                       // b*8+c, groups of 8 lanes per b
  float z = 2.0f * elog[t];
  float mx = z;
  for (int m2 = 1; m2 < 8; m2 <<= 1) mx = fmaxf(mx, __shfl_xor(mx, m2));
  float ex = __expf(z - mx);
  float sm = ex;
  for (int m2 = 1; m2 < 8; m2 <<= 1) sm += __shfl_xor(sm, m2);
  u[t] = ex / sm;
}

// ---------------- Pass 5: MVDR weights, one wave per (b,f) ----------------
// Gauss-Jordan on augmented [psd_n + eps*I | psd_s]; lane j owns column j (complex).
__global__ __launch_bounds__(256) void k_mvdr(const float* __restrict__ psd_s,
    const float* __restrict__ psd_n, const float* __restrict__ u,
    float* __restrict__ cbf) {
  int wv = threadIdx.x >> 5;
  int lane = threadIdx.x & 31;
  int bf = blockIdx.x * 8 + wv;
  if (bf >= B_ * F_) return;                 // wave-uniform
  int b = bf / F_;
  int j = lane & 15;
  int jc = j & 7;
  const float* src = (j < 8) ? psd_n : psd_s;
  float cr[8], ci[8];
  #pragma unroll
  for (int i = 0; i < 8; ++i) {
    const float* p = src + ((size_t)bf * 8 + i) * 16 + jc * 2;
    cr[i] = p[0]; ci[i] = p[1];
    if (j < 8 && i == jc) cr[i] += EPS_MVDR; // regularize diagonal of A
  }
  #pragma unroll
  for (int k = 0; k < 8; ++k) {
    float pkr[8], pki[8];
    #pragma unroll
    for (int i = 0; i < 8; ++i) { pkr[i] = __shfl(cr[i], k); pki[i] = __shfl(ci[i], k); }
    float d = 1.0f / (pkr[k] * pkr[k] + pki[k] * pki[k] + 1e-30f);
    float ipr = pkr[k] * d, ipi = -pki[k] * d;     // 1/pivot
    float nr = cr[k] * ipr - ci[k] * ipi;
    float ni = cr[k] * ipi + ci[k] * ipr;
    cr[k] = nr; ci[k] = ni;                        // scale row k
    #pragma unroll
    for (int i = 0; i < 8; ++i) {
      if (i == k) continue;
      cr[i] -= pkr[i] * nr - pki[i] * ni;          // eliminate
      ci[i] -= pkr[i] * ni + pki[i] * nr;
    }
  }
  // lanes 8..15 now hold X = inv(psd_n) @ psd_s columns
  float trr = 0.f, tri = 0.f;
  #pragma unroll
  for (int e = 0; e < 8; ++e) { trr += __shfl(cr[e], 8 + e); tri += __shfl(ci[e], 8 + e); }
  float t2 = trr + EPS_MVDR;
  float d = 1.0f / (t2 * t2 + tri * tri);
  float itr = t2 * d, iti = -tri * d;              // 1/(trace+eps)
  float uc = (lane >= 8 && lane < 16) ? u[b * 8 + jc] : 0.f;
  float pr[8], pim[8];
  #pragma unroll
  for (int e = 0; e < 8; ++e) { pr[e] = cr[e] * uc; pim[e] = ci[e] * uc; }
  #pragma unroll
  for (int e = 0; e < 8; ++e) {
    for (int m2 = 1; m2 < 8; m2 <<= 1) { pr[e] += __shfl_xor(pr[e], m2); pim[e] += __shfl_xor(pim[e], m2); }
  }
  if (lane == 8) {
    #pragma unroll
    for (int e = 0; e < 8; ++e) {
      float wr = pr[e] * itr - pim[e] * iti;
      float wi = pr[e] * iti + pim[e] * itr;
      cbf[(size_t)bf * 16 + e * 2]     = wr;       // store conj(bf)
      cbf[(size_t)bf * 16 + e * 2 + 1] = -wi;
    }
  }
}

// ---------------- Pass 6: apply beamformer, enh[b,t,f] ----------------
__global__ __launch_bounds__(256) void k_apply(const float* __restrict__ x,
    const float* __restrict__ cbf, float* __restrict__ out) {
  __shared__ float lb[F_ * 18];              // conj(bf) for this b, padded stride 18
  int b = blockIdx.y;
  int t0 = blockIdx.x * 32;
  for (int idx = threadIdx.x; idx < F_ * 16; idx += 256) {
    int f = idx >> 4, q = idx & 15;
    lb[f * 18 + q] = cbf[(size_t)(b * F_) * 16 + idx];
  }
  __syncthreads();
  for (int tt = 0; tt < 32; ++tt) {
    int t = t0 + tt;
    size_t xb = ((size_t)(b * T_ + t) * C_) * (F_ * 2);
    for (int f = threadIdx.x; f < F_; f += 256) {
      float ar = 0.f, ai = 0.f;
      #pragma unroll
      for (int c2 = 0; c2 < C_; ++c2) {
        float2 xv = *(const float2*)(x + xb + (size_t)c2 * (F_ * 2) + f * 2);
        float br = lb[f * 18 + c2 * 2], bi = lb[f * 18 + c2 * 2 + 1];
        ar += br * xv.x - bi * xv.y;
        ai += br * xv.y + bi * xv.x;
      }
      *(float2*)(out + ((size_t)(b * T_ + t) * F_ + f) * 2) = make_float2(ar, ai);
    }
  }
}

extern "C" void kernel_launch(void* const* d_in, const int* in_sizes, int n_in,
                              void* d_out, int out_size, void* d_ws, size_t ws_size,
                              hipStream_t stream) {
  const float* data = (const float*)d_in[0];   // (B,T,C,F,2)
  const float* ms   = (const float*)d_in[1];   // (B,F,C,T)
  const float* mn   = (const float*)d_in[2];   // (B,F,C,T)
  const float* W    = (const float*)d_in[3];   // (F,A)
  const float* bp   = (const float*)d_in[4];   // (A,)
  const float* wg   = (const float*)d_in[5];   // (A,)
  const float* bg   = (const float*)d_in[6];   // (1,)
  float* out = (float*)d_out;                  // (B,T,F,2)
  float* ws  = (float*)d_ws;
  float* msum  = ws;                           // B*F*2      = 8208
  float* psd_s = msum + 8208;                  // B*F*C*C*2  = 525312
  float* psd_n = psd_s + 525312;
  float* elog  = psd_n + 525312;               // 64
  float* u     = elog + 64;                    // 64
  float* cbf   = u + 64;                       // B*F*C*2    = 65664

  hipLaunchKernelGGL(k_mask_sums, dim3(B_ * F_), dim3(256), 0, stream, ms, mn, msum);
  hipLaunchKernelGGL(k_psd_wmma, dim3((F_ + FT - 1) / FT, B_), dim3(512), 0, stream,
                     data, ms, mn, msum, psd_s, psd_n);
  hipLaunchKernelGGL(k_attn, dim3(B_ * C_), dim3(320), 0, stream, psd_s, W, bp, wg, bg, elog);
  hipLaunchKernelGGL(k_softmax, dim3(1), dim3(64), 0, stream, elog, u);
  hipLaunchKernelGGL(k_mvdr, dim3((B_ * F_ + 7) / 8), dim3(256), 0, stream, psd_s, psd_n, u, cbf);
  hipLaunchKernelGGL(k_apply, dim3(T_ / 32, B_), dim3(256), 0, stream, data, cbf, out);
}